// FaceXZooProjector_2130303778910
// MI455X (gfx1250) — compile-verified
//
#include <hip/hip_runtime.h>
#include <stdint.h>

// Problem constants (fixed by the reference).
#define Bc    2
#define Hc    112
#define Wc    112
#define GRIDC 128
#define NVER  (GRIDC * GRIDC)   // 16384
#define NTRI  8192
#define HW    (Hc * Wc)         // 12544

// gfx1250 async global->LDS path (ASYNCcnt), guarded so compile is safe on
// any toolchain; fallback is a plain coalesced load.
#if defined(__has_builtin)
#if __has_builtin(__builtin_amdgcn_global_load_async_to_lds_b32)
#define HAVE_ASYNC_LDS 1
#endif
#endif
#ifndef HAVE_ASYNC_LDS
#define HAVE_ASYNC_LDS 0
#endif

typedef __attribute__((address_space(3))) int lds_i32;
typedef __attribute__((address_space(1))) int glb_i32;

// Map float to an order-preserving unsigned key (IEEE total order for finite values).
__device__ __forceinline__ unsigned int f2ord(float f) {
  unsigned int u = __float_as_uint(f);
  return (u & 0x80000000u) ? ~u : (u | 0x80000000u);
}

// One thread per triangle; loops over both batches reusing LDS-staged indices.
// Scatters 64-bit (depth | ~tri) keys into the per-pixel z-buffer with
// global_atomic_max_u64. Key semantics: larger depth wins; equal depth ->
// lower triangle index wins (matches jnp.argmax first-index tie-break).
__global__ void __launch_bounds__(256)
raster_kernel(const float* __restrict__ verts,          // [B,3,NVER]
              const int* __restrict__ tris,             // [3,NTRI]
              unsigned long long* __restrict__ keys) {  // [B,HW], pre-zeroed
  const int tidx = threadIdx.x;
  const int t = blockIdx.x * 256 + tidx;
  if (t >= NTRI) return;

  __shared__ int s_idx[3][256];

#if HAVE_ASYNC_LDS
  // Stage the three triangle-index planes through the async global->LDS path.
  // Each lane issues 3 async B32 copies (coalesced in memory), then waits on
  // ASYNCcnt before consuming its own staged entries.
  {
    for (int c = 0; c < 3; ++c) {
      const int* gp = tris + c * NTRI + t;
      int* lp = &s_idx[c][tidx];
      __builtin_amdgcn_global_load_async_to_lds_b32(
          (glb_i32*)(unsigned long long)gp,
          (lds_i32*)(unsigned int)(unsigned long long)lp,
          0, 0);
    }
#if defined(__has_builtin) && __has_builtin(__builtin_amdgcn_s_wait_asynccnt)
    __builtin_amdgcn_s_wait_asynccnt(0);
#else
    asm volatile("s_wait_asynccnt 0" ::: "memory");
#endif
  }
  const int t0 = s_idx[0][tidx];
  const int t1 = s_idx[1][tidx];
  const int t2 = s_idx[2][tidx];
#else
  const int t0 = tris[0 * NTRI + t];
  const int t1 = tris[1 * NTRI + t];
  const int t2 = tris[2 * NTRI + t];
  (void)s_idx;
#endif

  for (int b = 0; b < Bc; ++b) {
    const float* vx = verts + (size_t)b * 3 * NVER;  // x plane
    const float* vy = vx + NVER;                     // y plane
    const float* vz = vy + NVER;                     // z plane

    const float x0 = vx[t0], x1 = vx[t1], x2 = vx[t2];
    const float y0 = vy[t0], y1 = vy[t1], y2 = vy[t2];
    const float depth = (vz[t0] + vz[t1] + vz[t2]) / 3.0f;

    const float xmin = fminf(x0, fminf(x1, x2));
    const float xmax = fmaxf(x0, fmaxf(x1, x2));
    const float ymin = fminf(y0, fminf(y1, y2));
    const float ymax = fmaxf(y0, fmaxf(y1, y2));

    const int umin = (int)fmaxf(ceilf(xmin), 0.0f);
    const int umax = (int)fminf(floorf(xmax), (float)(Wc - 1));
    const int vmin = (int)fmaxf(ceilf(ymin), 0.0f);
    const int vmax = (int)fminf(floorf(ymax), (float)(Hc - 1));
    if (umin > umax || vmin > vmax) continue;  // empty clipped bbox

    const unsigned long long key =
        ((unsigned long long)f2ord(depth) << 32) |
        (unsigned long long)(0xFFFFFFFFu - (unsigned int)t);

    unsigned long long* kb = keys + (size_t)b * HW;
    for (int v = vmin; v <= vmax; ++v) {
      for (int u = umin; u <= umax; ++u) {
        atomicMax(&kb[v * Wc + u], key);  // global_atomic_max_u64
      }
    }
  }
}

// One thread per (b, pixel): decode winning triangle, recompute its flat color
// from the 9 corner-color gathers (L2-resident), write mask + 3 image planes.
__global__ void __launch_bounds__(256)
resolve_kernel(const float* __restrict__ colors,              // [B,3,NVER]
               const int* __restrict__ tris,                  // [3,NTRI]
               const unsigned long long* __restrict__ keys,   // [B,HW]
               float* __restrict__ out) {                     // mask(B*HW) ++ image(B*3*HW)
  const int idx = blockIdx.x * 256 + threadIdx.x;
  if (idx >= Bc * HW) return;
  const int b = idx / HW;
  const int p = idx - b * HW;

  // Confirmed gfx1250 path: lowers to global_prefetch_b8 (pulls the color
  // plane's line for this batch toward the WGP before the dependent gathers).
  __builtin_prefetch(colors + (size_t)b * 3 * NVER, 0, 0);

  const unsigned long long key = keys[idx];

  float r = 0.0f, g = 0.0f, bch = 0.0f, mask = 0.0f;
  if (key != 0ull) {
    const unsigned int tri = 0xFFFFFFFFu - (unsigned int)(key & 0xFFFFFFFFull);
    const int t0 = tris[tri];
    const int t1 = tris[NTRI + tri];
    const int t2 = tris[2 * NTRI + tri];
    const float* cb = colors + (size_t)b * 3 * NVER;
    r   = (cb[t0] + cb[t1] + cb[t2]) / 3.0f;
    g   = (cb[NVER + t0] + cb[NVER + t1] + cb[NVER + t2]) / 3.0f;
    bch = (cb[2 * NVER + t0] + cb[2 * NVER + t1] + cb[2 * NVER + t2]) / 3.0f;
    mask = 1.0f;
  }

  out[idx] = mask;  // face_mask [B,1,H,W] flattened == idx
  float* img = out + (size_t)Bc * HW;  // image [B,3,H,W]
  img[((size_t)b * 3 + 0) * HW + p] = r;
  img[((size_t)b * 3 + 1) * HW + p] = g;
  img[((size_t)b * 3 + 2) * HW + p] = bch;
}

extern "C" void kernel_launch(void* const* d_in, const int* in_sizes, int n_in,
                              void* d_out, int out_size, void* d_ws, size_t ws_size,
                              hipStream_t stream) {
  const float* verts  = (const float*)d_in[0];  // [B,3,NVER] f32
  const float* colors = (const float*)d_in[1];  // [B,3,NVER] f32
  const int*   tris   = (const int*)d_in[2];    // [3,NTRI]   i32
  // d_in[3]=h, d_in[4]=w are compile-time constants here (112, 112).
  (void)in_sizes; (void)n_in; (void)out_size; (void)ws_size;

  unsigned long long* keys = (unsigned long long*)d_ws;  // [B,HW] = 200,704 B

  // Zero the z-buffer keys (key==0 <=> no coverage). Graph-capturable.
  hipMemsetAsync(keys, 0, (size_t)Bc * HW * sizeof(unsigned long long), stream);

  raster_kernel<<<NTRI / 256, 256, 0, stream>>>(verts, tris, keys);

  const int npix = Bc * HW;
  resolve_kernel<<<(npix + 255) / 256, 256, 0, stream>>>(colors, tris, keys,
                                                         (float*)d_out);
}